// BlockDiagonal_76613626626441
// MI455X (gfx1250) — compile-verified
//
#include <hip/hip_runtime.h>
#include <stdint.h>
#include <math.h>

typedef __attribute__((ext_vector_type(2))) float        v2f;
typedef __attribute__((ext_vector_type(8))) float        v8f;
typedef __attribute__((ext_vector_type(4))) unsigned int v4u;
typedef __attribute__((ext_vector_type(8))) int          v8i;
typedef __attribute__((ext_vector_type(4))) int          v4i;

#define BATCH   8192
#define NFULL   4096
#define BS      512
#define NBLK    8

#define KSLAB     32          // K-slab per TDM transfer
#define SA_STRIDE 36          // 32 + 4 DW pad  (TDM pad: interval=32DW code 4, amount=4DW code 3)
#define SB_STRIDE 272         // 256 + 16 DW pad (TDM pad: interval=256DW code 7, amount=16DW code 15)

// ---------------------------------------------------------------------------
// Pass 1: effT[blk][k][n] = tanh(W[blk*512+n][blk*512+k])  -> 8 x 512 x 512 f32
// ---------------------------------------------------------------------------
__global__ __launch_bounds__(256) void tanh_transpose_kernel(const float* __restrict__ W,
                                                             float* __restrict__ effT) {
    int idx = blockIdx.x * 256 + threadIdx.x;
    int blk = idx >> 18;
    int rem = idx & ((1 << 18) - 1);
    int i   = rem >> 9;                 // K index within block
    int o   = rem & (BS - 1);           // N index within block (coalesced store)
    size_t w_idx = (size_t)(blk * BS + o) * NFULL + (size_t)(blk * BS + i);
    effT[idx] = tanhf(W[w_idx]);
}

// ---------------------------------------------------------------------------
// TDM: issue a 2D tensor_load_to_lds.  Descriptor per CDNA5 ISA.
// All dims/strides in elements (data_size = 4 bytes).
// ---------------------------------------------------------------------------
__device__ __forceinline__ void tdm_load_2d(unsigned int lds_off, const void* gptr,
                                            unsigned int tensor_d0, unsigned int tensor_d1,
                                            unsigned int stride0,
                                            unsigned int tile_d0,  unsigned int tile_d1,
                                            unsigned int pad_interval_code,
                                            unsigned int pad_amount_code) {
    uint64_t ga = (uint64_t)(uintptr_t)gptr;
    v4u g0;
    g0.x = 1u;                                                  // count=1, user descriptor
    g0.y = lds_off;                                             // LDS byte address
    g0.z = (unsigned int)(ga & 0xFFFFFFFFu);                    // global_addr[31:0]
    g0.w = (unsigned int)((ga >> 32) & 0x01FFFFFFu)             // global_addr[56:32]
         | 0x80000000u;                                         // type = 2 ("image")
    v8i g1;
    g1[0] = (int)((2u << 16)                                    // data_size = 4B
                | (1u << 20)                                    // pad_enable
                | (pad_interval_code << 22)
                | (pad_amount_code   << 25));
    g1[1] = (int)((tensor_d0 & 0xFFFFu) << 16);                 // tensor_dim0[15:0] @ bits 63:48
    g1[2] = (int)(((tensor_d0 >> 16) & 0xFFFFu)                 // tensor_dim0[31:16]
                | ((tensor_d1 & 0xFFFFu) << 16));               // tensor_dim1[15:0]
    g1[3] = (int)(((tensor_d1 >> 16) & 0xFFFFu)                 // tensor_dim1[31:16]
                | ((tile_d0 & 0xFFFFu) << 16));                 // tile_dim0
    g1[4] = (int)(tile_d1 & 0xFFFFu);                           // tile_dim1 (tile_dim2 = 0)
    g1[5] = (int)stride0;                                       // tensor_dim0_stride[31:0]
    g1[6] = 0;                                                  // stride0[47:32]=0, stride1 unused (2D)
    g1[7] = 0;
    v4i z4 = {0, 0, 0, 0};
#if __clang_major__ >= 23
    v8i z8 = {0, 0, 0, 0, 0, 0, 0, 0};
    __builtin_amdgcn_tensor_load_to_lds(g0, g1, z4, z4, z8, 0);
#else
    __builtin_amdgcn_tensor_load_to_lds(g0, g1, z4, z4, 0);
#endif
}

// ---------------------------------------------------------------------------
// Pass 2: block-diagonal GEMM.  TDM double-buffers A (64xKSLAB of x) and
// B (KSLABx256 of effT) into LDS; V_WMMA_F32_16X16X4_F32 consumes from LDS.
// Workgroup 256 threads = 8 waves (2M x 4N), wave tile 32M x 64N.
// Grid: x = 8192/64 = 128 M-tiles; y = 16 (8 blocks x 2 N-halves of 256).
// ---------------------------------------------------------------------------
__global__ __launch_bounds__(256, 2) void blockdiag_wmma_tdm_kernel(const float* __restrict__ x,
                                                                    const float* __restrict__ effT,
                                                                    float* __restrict__ out) {
    __shared__ float sA[2][64 * SA_STRIDE];      // 2 x 9 KB
    __shared__ float sB[2][KSLAB * SB_STRIDE];   // 2 x 34 KB

    const int tid    = threadIdx.x;
    const int wave   = tid >> 5;
    const int lane   = tid & 31;
    const int half   = lane >> 4;
    const int lanelo = lane & 15;
    const int wm     = wave & 1;                 // 2 waves along M
    const int wn     = wave >> 1;                // 4 waves along N

    const int blk   = blockIdx.y >> 1;                 // diagonal block
    const int nbase = (blockIdx.y & 1) * 256;          // N half within block
    const int m0    = blockIdx.x * 64;                 // WG M base

    const float* gA = x    + (size_t)m0 * NFULL + (size_t)blk * BS;   // [64 x 512] tile base
    const float* gB = effT + ((size_t)blk << 18) + nbase;             // [512 x 256] panel base

    v8f acc[2][4] = {};                               // [mt][nt] 16x16 f32 tiles

    const bool issuer = (wave == 0);
    if (issuer) {
        tdm_load_2d((unsigned int)(uintptr_t)&sA[0][0], gA,
                    NFULL, BATCH, NFULL, KSLAB, 64, /*pad*/ 4, 3);
        tdm_load_2d((unsigned int)(uintptr_t)&sB[0][0], gB,
                    BS, BS, BS, 256, KSLAB, /*pad*/ 7, 15);
        __builtin_amdgcn_s_wait_tensorcnt(0);
    }
    __syncthreads();

    for (int slab = 0; slab < BS / KSLAB; ++slab) {
        const int cur = slab & 1;
        const bool more = (slab + 1) < (BS / KSLAB);
        if (issuer && more) {
            tdm_load_2d((unsigned int)(uintptr_t)&sA[cur ^ 1][0],
                        gA + (size_t)(slab + 1) * KSLAB,
                        NFULL, BATCH, NFULL, KSLAB, 64, 4, 3);
            tdm_load_2d((unsigned int)(uintptr_t)&sB[cur ^ 1][0],
                        gB + (size_t)(slab + 1) * KSLAB * BS,
                        BS, BS, BS, 256, KSLAB, 7, 15);
        }

        const float* __restrict__ la = &sA[cur][0];
        const float* __restrict__ lb = &sB[cur][0];
#pragma unroll
        for (int k = 0; k < KSLAB; k += 4) {
            const int kk = k + 2 * half;                    // this lane's K pair base
            // A fragments (16x4): lane<16 K={k,k+1}, lane>=16 K={k+2,k+3}; M = lanelo
            v2f a0 = *(const v2f*)(la + (wm * 32 + lanelo)      * SA_STRIDE + kk);
            v2f a1 = *(const v2f*)(la + (wm * 32 + 16 + lanelo) * SA_STRIDE + kk);
            // B fragments (4x16): VGPR0 = row kk, VGPR1 = row kk+1; N = lanelo
            const float* bp = lb + kk * SB_STRIDE + wn * 64 + lanelo;
            v2f b[4];
#pragma unroll
            for (int u = 0; u < 4; ++u) {
                b[u].x = bp[u * 16];
                b[u].y = bp[u * 16 + SB_STRIDE];
            }
#pragma unroll
            for (int u = 0; u < 4; ++u) {
                acc[0][u] = __builtin_amdgcn_wmma_f32_16x16x4_f32(
                    false, a0, false, b[u], (short)0, acc[0][u], false, false);
                acc[1][u] = __builtin_amdgcn_wmma_f32_16x16x4_f32(
                    false, a1, false, b[u], (short)0, acc[1][u], false, false);
            }
        }

        if (issuer && more) __builtin_amdgcn_s_wait_tensorcnt(0);
        __syncthreads();
    }

    // D layout: VGPR r -> M = r + 8*half (+ subtile offsets); N = lanelo.
    const size_t col = (size_t)blk * BS + nbase + wn * 64 + lanelo;
#pragma unroll
    for (int mt = 0; mt < 2; ++mt) {
        float* orow = out + (size_t)(m0 + wm * 32 + mt * 16 + 8 * half) * NFULL + col;
#pragma unroll
        for (int u = 0; u < 4; ++u)
#pragma unroll
            for (int r = 0; r < 8; ++r)
                orow[(size_t)r * NFULL + u * 16] = acc[mt][u][r];
    }
}

extern "C" void kernel_launch(void* const* d_in, const int* in_sizes, int n_in,
                              void* d_out, int out_size, void* d_ws, size_t ws_size,
                              hipStream_t stream) {
    const float* x = (const float*)d_in[0];   // [8192, 4096] f32
    const float* W = (const float*)d_in[1];   // [4096, 4096] f32
    // d_in[2] (mask) unused: block-diagonal structure is static.
    float* out  = (float*)d_out;              // [8192, 4096] f32
    float* effT = (float*)d_ws;               // 8 * 512 * 512 f32 = 8 MB scratch

    {   // Pass 1: fused tanh + per-block transpose.
        int total = NBLK * BS * BS;
        tanh_transpose_kernel<<<total / 256, 256, 0, stream>>>(W, effT);
    }
    {   // Pass 2: TDM-fed block-diagonal WMMA GEMM.
        dim3 grid(BATCH / 64, NBLK * 2);      // 128 x 16
        blockdiag_wmma_tdm_kernel<<<grid, 256, 0, stream>>>(x, effT, out);
    }
}